// ChamferDistanceLoss_24215025614920
// MI455X (gfx1250) — compile-verified
//
#include <hip/hip_runtime.h>

// CDNA5 / gfx1250, wave32.
typedef __attribute__((ext_vector_type(2))) float v2f;
typedef __attribute__((ext_vector_type(8))) float v8f;

#define BATCH 8
#define NPTS 4096
#define TILES (NPTS / 16)                   // 256 column tiles of 16
#define WAVES_PER_BLOCK 8                   // 256 threads = 8 wave32
#define ROWBLOCKS_PER_BATCH (NPTS / 16)     // 256 row blocks of 16
#define BLOCKS_PER_BATCH (ROWBLOCKS_PER_BATCH / WAVES_PER_BLOCK)  // 32

// One directional Chamfer pass: for each point in S, min distance over T.
// Each wave owns a 16-row block of S and sweeps all 256 16-col tiles of T.
//
// Trick: V_WMMA_F32_16X16X4_F32 with
//   A' row m = (-2*sx, -2*sy, -2*sz, 1.0)
//   B' col n = ( tx,    ty,    tz,   t2 )   (t2 = ||t||^2, precomputed in LDS)
// gives acc[m][n] = t2 - 2<s,t>.  Since ||s||^2 is constant over the column
// sweep, we min acc directly and add ||s||^2 once at the end:
//   min_n d2 = ||s||^2 + min_n acc.
__global__ __launch_bounds__(256)
void chamfer_dir_kernel(const float* __restrict__ S, const float* __restrict__ T,
                        float* __restrict__ partial) {
    __shared__ float ldsT[NPTS * 4];   // 64 KB: padded (x,y,z,t2) per point

    const int batch    = blockIdx.x / BLOCKS_PER_BATCH;
    const int blockRow = blockIdx.x % BLOCKS_PER_BATCH;
    const float* Sb = S + (size_t)batch * NPTS * 3;
    const float* Tb = T + (size_t)batch * NPTS * 3;

    // Cooperative load of target batch into LDS, padding each point to
    // (x, y, z, ||t||^2).  16 points per thread, b128 LDS stores.
    for (int i = threadIdx.x; i < NPTS; i += blockDim.x) {
        const float x = Tb[i * 3 + 0];
        const float y = Tb[i * 3 + 1];
        const float z = Tb[i * 3 + 2];
        float4 v;
        v.x = x; v.y = y; v.z = z;
        v.w = fmaf(x, x, fmaf(y, y, z * z));
        ((float4*)ldsT)[i] = v;
    }
    __syncthreads();

    const int wave = threadIdx.x >> 5;
    const int lane = threadIdx.x & 31;
    const int half = lane >> 4;     // 0: K=0,1 ; 1: K=2,3
    const int l16  = lane & 15;
    const int rowblock = blockRow * WAVES_PER_BLOCK + wave;
    const int rowbase  = rowblock * 16;

    // A matrix (16x4 f32), pre-scaled by -2, K=3 slot = 1.0:
    //   lane<16 -> row l16 (-2x, -2y); lane>=16 -> row l16 (-2z, 1).
    v2f av;
    {
        const int r = rowbase + l16;
        if (half == 0) { av.x = -2.0f * Sb[r * 3 + 0]; av.y = -2.0f * Sb[r * 3 + 1]; }
        else           { av.x = -2.0f * Sb[r * 3 + 2]; av.y = 1.0f; }
    }

    // ||s||^2 for the 8 rows this lane-half covers in the C/D layout
    // (VGPR v holds row rowbase + v + 8*half).
    float s2h[8];
    {
        const int mbase = rowbase + half * 8;
        #pragma unroll
        for (int v = 0; v < 8; ++v) {
            const float x = Sb[(mbase + v) * 3 + 0];
            const float y = Sb[(mbase + v) * 3 + 1];
            const float z = Sb[(mbase + v) * 3 + 2];
            s2h[v] = fmaf(x, x, fmaf(y, y, z * z));
        }
    }

    float rmin[8];
    #pragma unroll
    for (int v = 0; v < 8; ++v) rmin[v] = 3.402823466e+38f;

    // Branchless B-operand address: column c = j*16 + l16, lane reads the
    // float2 at byte offset c*16 + half*8 -> (x,y) lower half, (z,t2) upper.
    // Word stride 4 across lanes => all 64 LDS banks hit exactly once.
    const float2* bp = (const float2*)ldsT + (l16 * 2 + half);

    #pragma unroll 4
    for (int j = 0; j < TILES; ++j) {
        const float2 b = bp[j * 32];
        v2f bv; bv.x = b.x; bv.y = b.y;
        v8f acc = {};
        // acc[m][n] = t2_n - 2*<s_m, t_n>
        acc = __builtin_amdgcn_wmma_f32_16x16x4_f32(
            /*neg_a=*/false, av, /*neg_b=*/false, bv,
            /*c_mod=*/(short)0, acc, /*reuse_a=*/false, /*reuse_b=*/false);
        #pragma unroll
        for (int v = 0; v < 8; ++v)
            rmin[v] = fminf(rmin[v], acc[v]);
    }

    // Min across the 16 lanes of each half (columns) -> per-row minima.
    #pragma unroll
    for (int m = 1; m < 16; m <<= 1) {
        #pragma unroll
        for (int v = 0; v < 8; ++v)
            rmin[v] = fminf(rmin[v], __shfl_xor(rmin[v], m, 32));
    }

    // d2 = ||s||^2 + min(acc); clamp + sqrt only after the min (monotonic).
    float sum = 0.0f;
    #pragma unroll
    for (int v = 0; v < 8; ++v)
        sum += sqrtf(fmaxf(rmin[v] + s2h[v], 0.0f));
    // Combine lower-half rows (0..7) and upper-half rows (8..15).
    sum += __shfl_xor(sum, 16, 32);

    if (lane == 0)
        partial[batch * ROWBLOCKS_PER_BATCH + rowblock] = sum;
}

// Deterministic fixed-order final reduction of all partials.
__global__ __launch_bounds__(256)
void chamfer_reduce_kernel(const float* __restrict__ partial, float* __restrict__ out,
                           int n, float scale) {
    __shared__ float sm[256];
    float s = 0.0f;
    for (int i = threadIdx.x; i < n; i += 256) s += partial[i];
    sm[threadIdx.x] = s;
    __syncthreads();
    for (int off = 128; off > 0; off >>= 1) {
        if ((int)threadIdx.x < off) sm[threadIdx.x] += sm[threadIdx.x + off];
        __syncthreads();
    }
    if (threadIdx.x == 0) out[0] = sm[0] * scale;
}

extern "C" void kernel_launch(void* const* d_in, const int* in_sizes, int n_in,
                              void* d_out, int out_size, void* d_ws, size_t ws_size,
                              hipStream_t stream) {
    const float* src = (const float*)d_in[0];   // [8,4096,3] f32
    const float* tgt = (const float*)d_in[1];   // [8,4096,3] f32
    float* out = (float*)d_out;                 // scalar f32
    float* ws  = (float*)d_ws;                  // >= 2*8*256 floats = 16 KB

    const dim3 grid(BATCH * BLOCKS_PER_BATCH);  // 256 blocks
    const dim3 block(256);                      // 8 wave32

    // term1: source -> nearest target
    chamfer_dir_kernel<<<grid, block, 0, stream>>>(src, tgt, ws);
    // term2: target -> nearest source
    chamfer_dir_kernel<<<grid, block, 0, stream>>>(tgt, src,
                                                   ws + BATCH * ROWBLOCKS_PER_BATCH);
    // out = (sum of all per-point min distances) / (B*N); N==M so one scale.
    chamfer_reduce_kernel<<<1, 256, 0, stream>>>(
        ws, out, 2 * BATCH * ROWBLOCKS_PER_BATCH, 1.0f / (BATCH * NPTS));
}